// ContentStyleReltLoss_81381040325479
// MI455X (gfx1250) — compile-verified
//
#include <hip/hip_runtime.h>
#include <math.h>

#define BATCH 4
#define CCH   256
#define HW    4096
#define TM    256      // output rows per block (2 m-blocks of 16 per wave, 8 waves)
#define TN    128      // output cols per block (8 n-subtiles of 16)
#define KC    16       // K per LDS stage
#define KPAIR 8        // KC/2
#define STA   544      // words per k-pair row, A tile: 2*TM+32  (mod 64 == 32)
#define STB   288      // words per k-pair row, B tile: 2*TN+32  (mod 64 == 32)
#define NSUB  8

typedef float v2f __attribute__((ext_vector_type(2)));
typedef float v8f __attribute__((ext_vector_type(8)));

__device__ __forceinline__ unsigned fkey(float f) {
  unsigned u = __float_as_uint(f);
  return (u & 0x80000000u) ? ~u : (u | 0x80000000u);
}
__device__ __forceinline__ float fdec(unsigned k) {
  unsigned u = (k & 0x80000000u) ? (k ^ 0x80000000u) : ~k;
  return __uint_as_float(u);
}
__device__ __forceinline__ void st_pair(float* p, float a, float b) {
  v2f t; t.x = a; t.y = b; *(v2f*)p = t;
}
__device__ __forceinline__ v2f ld_pair(const float* p) { return *(const v2f*)p; }

// ---------------- init ----------------
__global__ __launch_bounds__(256) void cs_init_kernel(unsigned* rowKey, unsigned* colKey, float* accum) {
  int i = blockIdx.x * 256 + threadIdx.x;
  if (i < BATCH * HW) { rowKey[i] = 0xFFFFFFFFu; colKey[i] = 0xFFFFFFFFu; }
  if (i == 0) accum[0] = 0.f;
}

// ---------------- 1/norm per column ----------------
__global__ __launch_bounds__(256) void cs_norm_kernel(const float* __restrict__ x, const float* __restrict__ cf,
                                                      const float* __restrict__ sf,
                                                      float* rnx, float* rnc, float* rns) {
  int p = blockIdx.x * 256 + threadIdx.x;
  const float* f; float* o;
  if (blockIdx.y == 0)      { f = x;  o = rnx; }
  else if (blockIdx.y == 1) { f = cf; o = rnc; }
  else                      { f = sf; o = rns; }
  int b = p / HW, i = p % HW;
  const float* base = f + (size_t)b * CCH * HW + i;
  float s = 0.f;
  #pragma unroll 4
  for (int c = 0; c < CCH; ++c) { float v = base[(size_t)c * HW]; s += v * v; }
  o[p] = rsqrtf(s);
}

// ---------------- w[b][c] = sum_i f[b,c,i] * rn[b,i] ----------------
__global__ __launch_bounds__(256) void cs_wvec_kernel(const float* __restrict__ x, const float* __restrict__ cf,
                                                      const float* __restrict__ rnx, const float* __restrict__ rnc,
                                                      float* wx, float* wc) {
  __shared__ float red[256];
  int bc = blockIdx.x;
  int b = bc / CCH, c = bc % CCH;
  const float* f  = (blockIdx.y == 0) ? x   : cf;
  const float* rn = (blockIdx.y == 0) ? rnx : rnc;
  float* w        = (blockIdx.y == 0) ? wx  : wc;
  const float* row = f + ((size_t)b * CCH + c) * HW;
  const float* rnb = rn + b * HW;
  float s = 0.f;
  for (int i = threadIdx.x; i < HW; i += 256) s += row[i] * rnb[i];
  red[threadIdx.x] = s; __syncthreads();
  for (int off = 128; off > 0; off >>= 1) {
    if (threadIdx.x < off) red[threadIdx.x] += red[threadIdx.x + off];
    __syncthreads();
  }
  if (threadIdx.x == 0) w[bc] = red[0];
}

// ---------------- 1/colsum:  colsum[b,j] = HW - dot(w_b, f[:,j]) * rn[b,j] ----------------
__global__ __launch_bounds__(256) void cs_colsum_kernel(const float* __restrict__ x, const float* __restrict__ cf,
                                                        const float* __restrict__ rnx, const float* __restrict__ rnc,
                                                        const float* __restrict__ wx, const float* __restrict__ wc,
                                                        float* rcsX, float* rcsY) {
  __shared__ float wsh[CCH];
  int p = blockIdx.x * 256 + threadIdx.x;
  int b = p / HW, j = p % HW;
  const float* f  = (blockIdx.y == 0) ? x    : cf;
  const float* rn = (blockIdx.y == 0) ? rnx  : rnc;
  const float* w  = (blockIdx.y == 0) ? wx   : wc;
  float* o        = (blockIdx.y == 0) ? rcsX : rcsY;
  wsh[threadIdx.x] = w[b * CCH + threadIdx.x];
  __syncthreads();
  const float* base = f + (size_t)b * CCH * HW + j;
  float dot = 0.f;
  #pragma unroll 4
  for (int c = 0; c < CCH; ++c) dot += wsh[c] * base[(size_t)c * HW];
  o[p] = 1.f / ((float)HW - dot * rn[p]);
}

// ---- global -> LDS, k-pair interleaved: word (k>>1)*stw + 2*col + (k&1) ----
__device__ __forceinline__ void load_tile_pair(float* lds, const float* g, int k0, int col0,
                                               int ncol, int stw, int t) {
  const int kp = t >> 5;             // 0..7 (wave index = k-pair)
  const int c4 = (t & 31) << 2;      // 0..124
  const float* r0p = g + (size_t)(k0 + 2 * kp)     * HW + col0;
  const float* r1p = g + (size_t)(k0 + 2 * kp + 1) * HW + col0;
  for (int s = 0; s < ncol; s += 128) {
    const int c = c4 + s;
    const float4 r0 = *(const float4*)(r0p + c);
    const float4 r1 = *(const float4*)(r1p + c);
    float* base = lds + kp * stw + c * 2;
    st_pair(base + 0, r0.x, r1.x);
    st_pair(base + 2, r0.y, r1.y);
    st_pair(base + 4, r0.z, r1.z);
    st_pair(base + 6, r0.w, r1.w);
  }
}

// ---------------- fused content pass: Gx = x'.x , Gy = cf'.cf ----------------
__global__ __launch_bounds__(256) void cs_content_kernel(const float* __restrict__ x, const float* __restrict__ cf,
                                                         const float* __restrict__ rnx, const float* __restrict__ rnc,
                                                         const float* __restrict__ rcsX, const float* __restrict__ rcsY,
                                                         float* __restrict__ accum) {
  __shared__ float ldsXi[KPAIR * STA], ldsYi[KPAIR * STA];
  __shared__ float ldsXj[KPAIR * STB], ldsYj[KPAIR * STB];
  __shared__ float red[256];
  const int b  = blockIdx.z;
  const int i0 = blockIdx.y * TM;
  const int j0 = blockIdx.x * TN;
  const int t    = threadIdx.x;
  const int lane = t & 31;
  const int w    = t >> 5;
  const int m0   = w << 5;            // wave's 32-row block
  const int mc   = lane & 15;
  const int hi   = lane >> 4;         // upper half-wave -> k-pair +1, M +8

  const float* xb = x  + (size_t)b * CCH * HW;
  const float* yb = cf + (size_t)b * CCH * HW;

  v8f accX[2][NSUB], accY[2][NSUB];
  #pragma unroll
  for (int q = 0; q < 2; ++q)
    #pragma unroll
    for (int n = 0; n < NSUB; ++n)
      #pragma unroll
      for (int e = 0; e < 8; ++e) { accX[q][n][e] = 0.f; accY[q][n][e] = 0.f; }

  for (int k0 = 0; k0 < CCH; k0 += KC) {
    __syncthreads();
    load_tile_pair(ldsXi, xb, k0, i0, TM, STA, t);
    load_tile_pair(ldsYi, yb, k0, i0, TM, STA, t);
    load_tile_pair(ldsXj, xb, k0, j0, TN, STB, t);
    load_tile_pair(ldsYj, yb, k0, j0, TN, STB, t);
    __syncthreads();
    #pragma unroll
    for (int kk = 0; kk < KC; kk += 4) {
      const int arow = ((kk >> 1) + hi) * STA;
      const int brow = ((kk >> 1) + hi) * STB;
      v2f aX[2], aY[2];
      aX[0] = ld_pair(ldsXi + arow + ((m0      + mc) << 1));
      aX[1] = ld_pair(ldsXi + arow + ((m0 + 16 + mc) << 1));
      aY[0] = ld_pair(ldsYi + arow + ((m0      + mc) << 1));
      aY[1] = ld_pair(ldsYi + arow + ((m0 + 16 + mc) << 1));
      #pragma unroll
      for (int n = 0; n < NSUB; ++n) {
        const v2f bX = ld_pair(ldsXj + brow + (((n << 4) + mc) << 1));
        const v2f bY = ld_pair(ldsYj + brow + (((n << 4) + mc) << 1));
        accX[0][n] = __builtin_amdgcn_wmma_f32_16x16x4_f32(false, aX[0], false, bX, (short)0, accX[0][n], false, false);
        accX[1][n] = __builtin_amdgcn_wmma_f32_16x16x4_f32(false, aX[1], false, bX, (short)0, accX[1][n], false, false);
        accY[0][n] = __builtin_amdgcn_wmma_f32_16x16x4_f32(false, aY[0], false, bY, (short)0, accY[0][n], false, false);
        accY[1][n] = __builtin_amdgcn_wmma_f32_16x16x4_f32(false, aY[1], false, bY, (short)0, accY[1][n], false, false);
      }
    }
  }

  // epilogue: C/D layout — VGPR r -> M = r (+8 upper half), N = lane%16
  const float* rnxb = rnx + b * HW;
  const float* rncb = rnc + b * HW;
  float rxi[2][8], ryi[2][8];
  #pragma unroll
  for (int q = 0; q < 2; ++q) {
    const int ibq = i0 + m0 + (q << 4) + (hi << 3);
    #pragma unroll
    for (int r = 0; r < 8; ++r) { rxi[q][r] = rnxb[ibq + r]; ryi[q][r] = rncb[ibq + r]; }
  }

  float s = 0.f;
  #pragma unroll
  for (int n = 0; n < NSUB; ++n) {
    const int j = j0 + (n << 4) + mc;
    const float rxj = rnxb[j], ryj = rncb[j];
    const float cxj = rcsX[b * HW + j], cyj = rcsY[b * HW + j];
    #pragma unroll
    for (int q = 0; q < 2; ++q)
      #pragma unroll
      for (int r = 0; r < 8; ++r) {
        const float mx = 1.f - accX[q][n][r] * rxi[q][r] * rxj;
        const float my = 1.f - accY[q][n][r] * ryi[q][r] * ryj;
        s += fabsf(mx * cxj - my * cyj);
      }
  }
  red[t] = s; __syncthreads();
  for (int off = 128; off > 0; off >>= 1) {
    if (t < off) red[t] += red[t + off];
    __syncthreads();
  }
  if (t == 0) atomicAdd(accum, red[0]);
}

// ---------------- style pass: G = x'.sf with fused row/col mins ----------------
__global__ __launch_bounds__(256) void cs_style_kernel(const float* __restrict__ x, const float* __restrict__ sf,
                                                       const float* __restrict__ rnx, const float* __restrict__ rns,
                                                       unsigned* __restrict__ rowKey, unsigned* __restrict__ colKey) {
  __shared__ float ldsXi[KPAIR * STA], ldsSj[KPAIR * STB];
  __shared__ unsigned srow[TM], scol[TN];
  const int b  = blockIdx.z;
  const int i0 = blockIdx.y * TM;
  const int j0 = blockIdx.x * TN;
  const int t    = threadIdx.x;
  const int lane = t & 31;
  const int w    = t >> 5;
  const int m0   = w << 5;
  const int mc   = lane & 15;
  const int hi   = lane >> 4;

  srow[t] = 0xFFFFFFFFu;
  if (t < TN) scol[t] = 0xFFFFFFFFu;

  const float* xb = x  + (size_t)b * CCH * HW;
  const float* sb = sf + (size_t)b * CCH * HW;

  v8f acc[2][NSUB];
  #pragma unroll
  for (int q = 0; q < 2; ++q)
    #pragma unroll
    for (int n = 0; n < NSUB; ++n)
      #pragma unroll
      for (int e = 0; e < 8; ++e) acc[q][n][e] = 0.f;

  for (int k0 = 0; k0 < CCH; k0 += KC) {
    __syncthreads();
    load_tile_pair(ldsXi, xb, k0, i0, TM, STA, t);
    load_tile_pair(ldsSj, sb, k0, j0, TN, STB, t);
    __syncthreads();
    #pragma unroll
    for (int kk = 0; kk < KC; kk += 4) {
      const int arow = ((kk >> 1) + hi) * STA;
      const int brow = ((kk >> 1) + hi) * STB;
      v2f a0 = ld_pair(ldsXi + arow + ((m0      + mc) << 1));
      v2f a1 = ld_pair(ldsXi + arow + ((m0 + 16 + mc) << 1));
      #pragma unroll
      for (int n = 0; n < NSUB; ++n) {
        const v2f bv = ld_pair(ldsSj + brow + (((n << 4) + mc) << 1));
        acc[0][n] = __builtin_amdgcn_wmma_f32_16x16x4_f32(false, a0, false, bv, (short)0, acc[0][n], false, false);
        acc[1][n] = __builtin_amdgcn_wmma_f32_16x16x4_f32(false, a1, false, bv, (short)0, acc[1][n], false, false);
      }
    }
  }

  const float* rnxb = rnx + b * HW;
  const float* rnsb = rns + b * HW;
  float rsj[NSUB], colm[NSUB];
  #pragma unroll
  for (int n = 0; n < NSUB; ++n) { rsj[n] = rnsb[j0 + (n << 4) + mc]; colm[n] = 3.0e38f; }

  #pragma unroll
  for (int q = 0; q < 2; ++q) {
    const int ibq = m0 + (q << 4) + (hi << 3);   // local row base
    #pragma unroll
    for (int r = 0; r < 8; ++r) {
      float rowm = 3.0e38f;
      const float rxi = rnxb[i0 + ibq + r];
      #pragma unroll
      for (int n = 0; n < NSUB; ++n) {
        float d = 1.f - acc[q][n][r] * rxi * rsj[n];
        rowm = fminf(rowm, d);
        colm[n] = fminf(colm[n], d);
      }
      rowm = fminf(rowm, __shfl_xor(rowm, 1, 32));
      rowm = fminf(rowm, __shfl_xor(rowm, 2, 32));
      rowm = fminf(rowm, __shfl_xor(rowm, 4, 32));
      rowm = fminf(rowm, __shfl_xor(rowm, 8, 32));
      if (mc == 0) atomicMin(&srow[ibq + r], fkey(rowm));
    }
  }
  #pragma unroll
  for (int n = 0; n < NSUB; ++n) {
    float cm = fminf(colm[n], __shfl_xor(colm[n], 16, 32));
    if (lane < 16) atomicMin(&scol[(n << 4) + lane], fkey(cm));
  }
  __syncthreads();
  atomicMin(&rowKey[b * HW + i0 + t], srow[t]);
  if (t < TN) atomicMin(&colKey[b * HW + j0 + t], scol[t]);
}

// ---------------- finalize ----------------
__global__ __launch_bounds__(256) void cs_final_kernel(const float* __restrict__ accum,
                                                       const unsigned* __restrict__ rowKey,
                                                       const unsigned* __restrict__ colKey,
                                                       float* __restrict__ out) {
  __shared__ float r1[256], r2[256];
  float s1 = 0.f, s2 = 0.f;
  for (int i = threadIdx.x; i < BATCH * HW; i += 256) { s1 += fdec(rowKey[i]); s2 += fdec(colKey[i]); }
  r1[threadIdx.x] = s1; r2[threadIdx.x] = s2; __syncthreads();
  for (int off = 128; off > 0; off >>= 1) {
    if (threadIdx.x < off) { r1[threadIdx.x] += r1[threadIdx.x + off]; r2[threadIdx.x] += r2[threadIdx.x + off]; }
    __syncthreads();
  }
  if (threadIdx.x == 0) {
    const float inv = 1.f / (float)(BATCH * HW);
    out[0] = accum[0] * inv;                       // mean(|..|)*HW
    out[1] = fmaxf(r1[0] * inv, r2[0] * inv);
  }
}

extern "C" void kernel_launch(void* const* d_in, const int* in_sizes, int n_in,
                              void* d_out, int out_size, void* d_ws, size_t ws_size,
                              hipStream_t stream) {
  const float* x  = (const float*)d_in[0];
  const float* cf = (const float*)d_in[1];
  const float* sf = (const float*)d_in[2];
  float* out = (float*)d_out;

  char* p = (char*)d_ws;
  float*    rnx    = (float*)p;    p += (size_t)BATCH * HW * sizeof(float);
  float*    rnc    = (float*)p;    p += (size_t)BATCH * HW * sizeof(float);
  float*    rns    = (float*)p;    p += (size_t)BATCH * HW * sizeof(float);
  float*    rcsX   = (float*)p;    p += (size_t)BATCH * HW * sizeof(float);
  float*    rcsY   = (float*)p;    p += (size_t)BATCH * HW * sizeof(float);
  float*    wx     = (float*)p;    p += (size_t)BATCH * CCH * sizeof(float);
  float*    wc     = (float*)p;    p += (size_t)BATCH * CCH * sizeof(float);
  unsigned* rowKey = (unsigned*)p; p += (size_t)BATCH * HW * sizeof(unsigned);
  unsigned* colKey = (unsigned*)p; p += (size_t)BATCH * HW * sizeof(unsigned);
  float*    accum  = (float*)p;

  cs_init_kernel<<<(BATCH * HW + 255) / 256, 256, 0, stream>>>(rowKey, colKey, accum);

  dim3 gN(BATCH * HW / 256, 3);
  cs_norm_kernel<<<gN, 256, 0, stream>>>(x, cf, sf, rnx, rnc, rns);

  dim3 gW(BATCH * CCH, 2);
  cs_wvec_kernel<<<gW, 256, 0, stream>>>(x, cf, rnx, rnc, wx, wc);

  dim3 gS(BATCH * HW / 256, 2);
  cs_colsum_kernel<<<gS, 256, 0, stream>>>(x, cf, rnx, rnc, wx, wc, rcsX, rcsY);

  dim3 gT(HW / TN, HW / TM, BATCH);
  cs_content_kernel<<<gT, 256, 0, stream>>>(x, cf, rnx, rnc, rcsX, rcsY, accum);
  cs_style_kernel<<<gT, 256, 0, stream>>>(x, sf, rnx, rns, rowKey, colKey);

  cs_final_kernel<<<1, 256, 0, stream>>>(accum, rowKey, colKey, out);
}